// LIF_40329742909559
// MI455X (gfx1250) — compile-verified
//
#include <hip/hip_runtime.h>

// Persistent fused SNN forward for MI455X (gfx1250, wave32, WMMA).
// One kernel launch runs all T=100 timesteps; each block owns 16 batch rows,
// keeps m1..m5 in VGPRs (WMMA D-layout), stages activations in padded LDS,
// and does all GEMMs with V_WMMA_F32_16X16X4_F32 (exact fp32).

typedef __attribute__((ext_vector_type(2))) float v2f;
typedef __attribute__((ext_vector_type(8))) float v8f;

#define T_STEPS 100
#define BATCH   2048
#define HID     512
#define OUTF    128
#define MT      16          // batch rows per block (one WMMA M tile)
#define ASTRIDE 516         // padded LDS row stride (floats) -> conflict-free A frags
#define KT_H    (HID / 4)   // 128 K-steps of 4 for K=512

// packed-weight segment offsets (in v2f units): per matrix, layout is
// [(ntile*KT_H + kt)*32 + lane] -> one contiguous 256B block per B-fragment.
#define PK_W2 0
#define PK_W3 (32 * KT_H * 32)          // 131072
#define PK_W4 (2 * 32 * KT_H * 32)      // 262144
#define PK_W5 (3 * 32 * KT_H * 32)      // 393216
#define PK_TOTAL (PK_W5 + 8 * KT_H * 32) // 425984 v2f = 3.25 MB

__device__ __forceinline__ float clamp01(float v) {
    return fminf(fmaxf(v, 0.0f), 1.0f);
}

__device__ __forceinline__ v8f wmma4(v2f a, v2f b, v8f c) {
    // D = A(16x4,f32) * B(4x16,f32) + C(16x16,f32)
    return __builtin_amdgcn_wmma_f32_16x16x4_f32(
        false, a, false, b, (short)0, c, false, false);
}

// ---------------------------------------------------------------------------
// One-time weight repack into WMMA B-fragment order.
// B(4x16) fragment for (ntile, kt): lane -> n = ntile*16 + lane%16,
// k = kt*4 + 2*(lane/16); value pair {W[n][k], W[n][k+1]}.
// ---------------------------------------------------------------------------
__global__ void repack_weights(const float* __restrict__ W2,
                               const float* __restrict__ W3,
                               const float* __restrict__ W4,
                               const float* __restrict__ W5,
                               v2f* __restrict__ out) {
    int idx = blockIdx.x * 256 + threadIdx.x;
    if (idx >= PK_TOTAL) return;
    const float* W;
    int base;
    if (idx < PK_W3)       { W = W2; base = PK_W2; }
    else if (idx < PK_W4)  { W = W3; base = PK_W3; }
    else if (idx < PK_W5)  { W = W4; base = PK_W4; }
    else                   { W = W5; base = PK_W5; }
    int r     = idx - base;
    int lane  = r & 31;
    int kt    = (r >> 5) & (KT_H - 1);
    int ntile = r >> 12;                  // r / (KT_H*32)
    int n = ntile * 16 + (lane & 15);
    int k = kt * 4 + ((lane >> 4) << 1);
    v2f v;
    v.x = W[(size_t)n * HID + k];
    v.y = W[(size_t)n * HID + k + 1];
    out[idx] = v;
}

// ---------------------------------------------------------------------------
// GEMM over K=512 for 4 adjacent N-tiles (H-wide layers). A from LDS.
// ---------------------------------------------------------------------------
template <bool PACKED>
__device__ __forceinline__ void gemm_h(const float* __restrict__ act,
                                       int l16, int khalf, int lane, int w,
                                       const float* __restrict__ Wg,
                                       const v2f* __restrict__ pk,
                                       v8f acc[4], const float bias[4]) {
    #pragma unroll
    for (int j = 0; j < 4; ++j) {
        v8f a;
        #pragma unroll
        for (int v = 0; v < 8; ++v) a[v] = bias[j];
        acc[j] = a;
    }
    const float* arow = act + l16 * ASTRIDE + khalf;
    const v2f* p[4];
    const float* wr[4];
    #pragma unroll
    for (int j = 0; j < 4; ++j) {
        int nt = w * 4 + j;
        if constexpr (PACKED) {
            p[j] = pk + ((size_t)nt * KT_H * 32 + lane);
        } else {
            wr[j] = Wg + ((size_t)(nt * 16 + l16) * HID + khalf);
        }
    }
    #pragma unroll 4
    for (int kt = 0; kt < KT_H; ++kt) {
        v2f a = *(const v2f*)(arow + kt * 4);
        #pragma unroll
        for (int j = 0; j < 4; ++j) {
            v2f bb;
            if constexpr (PACKED) bb = p[j][kt * 32];
            else                  bb = *(const v2f*)(wr[j] + kt * 4);
            acc[j] = wmma4(a, bb, acc[j]);
        }
    }
}

// Single-N-tile GEMM for the O=128 layer (ntile == wave id).
template <bool PACKED>
__device__ __forceinline__ v8f gemm_o(const float* __restrict__ act,
                                      int l16, int khalf, int lane, int w,
                                      const float* __restrict__ Wg,
                                      const v2f* __restrict__ pk, float bias) {
    v8f acc;
    #pragma unroll
    for (int v = 0; v < 8; ++v) acc[v] = bias;
    const float* arow = act + l16 * ASTRIDE + khalf;
    const v2f* p = nullptr;
    const float* wr = nullptr;
    if constexpr (PACKED) p  = pk + ((size_t)w * KT_H * 32 + lane);
    else                  wr = Wg + ((size_t)(w * 16 + l16) * HID + khalf);
    #pragma unroll 4
    for (int kt = 0; kt < KT_H; ++kt) {
        v2f a = *(const v2f*)(arow + kt * 4);
        v2f bb;
        if constexpr (PACKED) bb = p[kt * 32];
        else                  bb = *(const v2f*)(wr + kt * 4);
        acc = wmma4(a, bb, acc);
    }
    return acc;
}

// ---------------------------------------------------------------------------
// Persistent fused kernel: 128 blocks x 256 threads (8 wave32).
// ---------------------------------------------------------------------------
template <bool PACKED>
__global__ __launch_bounds__(256) void snn_fused(
    const float* __restrict__ x,
    const float* __restrict__ W1, const float* __restrict__ b1,
    const float* __restrict__ beta1, const float* __restrict__ thr1,
    const float* __restrict__ W2, const float* __restrict__ b2,
    const float* __restrict__ beta2, const float* __restrict__ thr2,
    const float* __restrict__ W3, const float* __restrict__ b3,
    const float* __restrict__ beta3, const float* __restrict__ thr3,
    const float* __restrict__ W4, const float* __restrict__ b4,
    const float* __restrict__ beta4, const float* __restrict__ thr4,
    const float* __restrict__ W5, const float* __restrict__ b5,
    const float* __restrict__ beta5, const float* __restrict__ thr5,
    const v2f* __restrict__ pk,
    float* __restrict__ cur_rec, float* __restrict__ mem_rec,
    float* __restrict__ spk_rec) {
    __shared__ float act[MT * ASTRIDE];
    __shared__ float xr[MT];
    __shared__ float rowsum[MT];

    const int tid   = threadIdx.x;
    const int w     = tid >> 5;
    const int lane  = tid & 31;
    const int l16   = lane & 15;
    const int half  = lane >> 4;
    const int khalf = half << 1;
    const int b0    = blockIdx.x * MT;

    // ---- loop-invariant per-lane parameters ----
    const int c1a = w * 64 + lane, c1b = c1a + 32;
    float w1v[2] = {W1[c1a], W1[c1b]};
    float b1v[2] = {b1[c1a], b1[c1b]};
    float cb1[2] = {clamp01(beta1[c1a]), clamp01(beta1[c1b])};
    float th1[2] = {thr1[c1a], thr1[c1b]};

    float b2c[4], cb2[4], th2[4], b3c[4], cb3[4], th3[4], b4c[4], cb4[4];
    #pragma unroll
    for (int j = 0; j < 4; ++j) {
        int col = w * 64 + j * 16 + l16;
        b2c[j] = b2[col]; cb2[j] = clamp01(beta2[col]); th2[j] = thr2[col];
        b3c[j] = b3[col]; cb3[j] = clamp01(beta3[col]); th3[j] = thr3[col];
        b4c[j] = b4[col]; cb4[j] = clamp01(beta4[col]);
    }
    const int col5 = w * 16 + l16;
    const float b5c = b5[col5], cb5 = clamp01(beta5[col5]);

    // ---- persistent state ----
    float m1[32];
    #pragma unroll
    for (int i = 0; i < 32; ++i) m1[i] = 0.0f;
    v8f m2s[4] = {}, m3s[4] = {}, m4s[4] = {};
    v8f m5s = {};

    for (int t = 0; t < T_STEPS; ++t) {
        if (tid < MT) {
            xr[tid]     = x[(size_t)t * BATCH + b0 + tid];
            rowsum[tid] = 0.0f;
        }
        __syncthreads();

        // ---- layer 1: outer-product current + LIF(sub) -> s1 in LDS ----
        #pragma unroll
        for (int g = 0; g < 2; ++g) {
            const int col = (g == 0) ? c1a : c1b;
            #pragma unroll
            for (int m = 0; m < MT; ++m) {
                float cc = xr[m] * w1v[g] + b1v[g];
                float mo = m1[g * 16 + m];
                float rr = (mo - th1[g] > 0.0f) ? th1[g] : 0.0f;
                float mn = cb1[g] * mo + cc - rr;
                m1[g * 16 + m] = mn;
                act[m * ASTRIDE + col] = (mn - th1[g] > 0.0f) ? 1.0f : 0.0f;
            }
        }
        __syncthreads();

        // ---- layer 2: s1 @ W2^T + b2 ; LIF(sub) -> s2 in LDS ----
        v8f acc[4];
        gemm_h<PACKED>(act, l16, khalf, lane, w, W2, pk + PK_W2, acc, b2c);
        __syncthreads();                    // everyone done reading s1
        #pragma unroll
        for (int j = 0; j < 4; ++j) {
            int col = w * 64 + j * 16 + l16;
            #pragma unroll
            for (int v = 0; v < 8; ++v) {
                int mr = v + (half << 3);
                float mo = m2s[j][v];
                float rr = (mo - th2[j] > 0.0f) ? th2[j] : 0.0f;
                float mn = cb2[j] * mo + acc[j][v] - rr;
                m2s[j][v] = mn;
                act[mr * ASTRIDE + col] = (mn - th2[j] > 0.0f) ? 1.0f : 0.0f;
            }
        }
        __syncthreads();

        // ---- layer 3: s2 @ W3^T + b3 ; LIF(sub) -> m3 values to LDS, s3 out ----
        gemm_h<PACKED>(act, l16, khalf, lane, w, W3, pk + PK_W3, acc, b3c);
        __syncthreads();
        #pragma unroll
        for (int j = 0; j < 4; ++j) {
            int col = w * 64 + j * 16 + l16;
            #pragma unroll
            for (int v = 0; v < 8; ++v) {
                int mr = v + (half << 3);
                float mo = m3s[j][v];
                float rr = (mo - th3[j] > 0.0f) ? th3[j] : 0.0f;
                float mn = cb3[j] * mo + acc[j][v] - rr;
                m3s[j][v] = mn;
                act[mr * ASTRIDE + col] = mn;   // layer 4 consumes membrane, not spikes
                spk_rec[((size_t)t * BATCH + b0 + mr) * HID + col] =
                    (mn - th3[j] > 0.0f) ? 1.0f : 0.0f;
            }
        }
        __syncthreads();

        // ---- layer 4: m3 @ W4^T + b4 ; LIF(none) -> m4 values to LDS ----
        gemm_h<PACKED>(act, l16, khalf, lane, w, W4, pk + PK_W4, acc, b4c);
        __syncthreads();
        #pragma unroll
        for (int j = 0; j < 4; ++j) {
            int col = w * 64 + j * 16 + l16;
            #pragma unroll
            for (int v = 0; v < 8; ++v) {
                int mr = v + (half << 3);
                float mn = cb4[j] * m4s[j][v] + acc[j][v];
                m4s[j][v] = mn;
                act[mr * ASTRIDE + col] = mn;   // layer 5 consumes membrane
            }
        }
        __syncthreads();

        // ---- layer 5: m4 @ W5^T + b5 ; LIF(none); outputs c5 and mean(m5) ----
        v8f c5 = gemm_o<PACKED>(act, l16, khalf, lane, w, W5, pk + PK_W5, b5c);
        #pragma unroll
        for (int v = 0; v < 8; ++v) {
            int mr = v + (half << 3);
            cur_rec[((size_t)t * BATCH + b0 + mr) * OUTF + col5] = c5[v];
            float mn = cb5 * m5s[v] + c5[v];
            m5s[v] = mn;
            atomicAdd(&rowsum[mr], mn);
        }
        __syncthreads();
        if (tid < MT) {
            mem_rec[(size_t)t * BATCH + b0 + tid] = rowsum[tid] * (1.0f / OUTF);
        }
        // next-iteration xr/rowsum rewrite is by the same tid<MT threads,
        // followed by __syncthreads(), so no extra barrier needed here.
    }
}

// ---------------------------------------------------------------------------
extern "C" void kernel_launch(void* const* d_in, const int* in_sizes, int n_in,
                              void* d_out, int out_size, void* d_ws,
                              size_t ws_size, hipStream_t stream) {
    const float* x     = (const float*)d_in[0];
    const float* W1    = (const float*)d_in[1];
    const float* b1    = (const float*)d_in[2];
    const float* beta1 = (const float*)d_in[3];
    const float* thr1  = (const float*)d_in[4];
    const float* W2    = (const float*)d_in[5];
    const float* b2    = (const float*)d_in[6];
    const float* beta2 = (const float*)d_in[7];
    const float* thr2  = (const float*)d_in[8];
    const float* W3    = (const float*)d_in[9];
    const float* b3    = (const float*)d_in[10];
    const float* beta3 = (const float*)d_in[11];
    const float* thr3  = (const float*)d_in[12];
    const float* W4    = (const float*)d_in[13];
    const float* b4    = (const float*)d_in[14];
    const float* beta4 = (const float*)d_in[15];
    const float* thr4  = (const float*)d_in[16];
    const float* W5    = (const float*)d_in[17];
    const float* b5    = (const float*)d_in[18];
    const float* beta5 = (const float*)d_in[19];
    const float* thr5  = (const float*)d_in[20];

    float* out = (float*)d_out;
    float* cur = out;
    float* mem = cur + (size_t)T_STEPS * BATCH * OUTF;
    float* spk = mem + (size_t)T_STEPS * BATCH;

    const size_t pk_bytes = (size_t)PK_TOTAL * sizeof(v2f);
    const bool packed = (d_ws != nullptr) && (ws_size >= pk_bytes);

    dim3 grid(BATCH / MT), block(256);
    if (packed) {
        repack_weights<<<(PK_TOTAL + 255) / 256, 256, 0, stream>>>(
            W2, W3, W4, W5, (v2f*)d_ws);
        snn_fused<true><<<grid, block, 0, stream>>>(
            x, W1, b1, beta1, thr1, W2, b2, beta2, thr2, W3, b3, beta3, thr3,
            W4, b4, beta4, thr4, W5, b5, beta5, thr5, (const v2f*)d_ws,
            cur, mem, spk);
    } else {
        snn_fused<false><<<grid, block, 0, stream>>>(
            x, W1, b1, beta1, thr1, W2, b2, beta2, thr2, W3, b3, beta3, thr3,
            W4, b4, beta4, thr4, W5, b5, beta5, thr5, nullptr,
            cur, mem, spk);
    }
    (void)in_sizes; (void)n_in; (void)out_size; (void)thr4; (void)thr5;
}